// MetaPathGNN_40535901339973
// MI455X (gfx1250) — compile-verified
//
#include <hip/hip_runtime.h>

#define H 128

typedef __attribute__((ext_vector_type(2))) float v2f;
typedef __attribute__((ext_vector_type(8))) float v8f;

// ---------------------------------------------------------------------------
// Fold W01 = conv1_w0_w + conv1_w1_w ; bias_h = wl_b + w0_b + w1_b
// ---------------------------------------------------------------------------
__global__ void metapath_prep_kernel(const float* __restrict__ w0w,
                                     const float* __restrict__ w1w,
                                     const float* __restrict__ w0b,
                                     const float* __restrict__ wlb,
                                     const float* __restrict__ w1b,
                                     float* __restrict__ W01,
                                     float* __restrict__ biasH) {
    int i = blockIdx.x * blockDim.x + threadIdx.x;
    if (i < H * H) W01[i] = w0w[i] + w1w[i];
    if (i < H)     biasH[i] = wlb[i] + w0b[i] + w1b[i];
}

// ---------------------------------------------------------------------------
// Zero the aggregation buffer (graph-capture safe)
// ---------------------------------------------------------------------------
__global__ void metapath_zero_kernel(float4* __restrict__ p, int n4) {
    int i = blockIdx.x * blockDim.x + threadIdx.x;
    if (i < n4) p[i] = make_float4(0.f, 0.f, 0.f, 0.f);
}

// ---------------------------------------------------------------------------
// One wave per edge: coalesced 512B row gather from x_a, 4 f32 atomics/lane
// into aggr[dst]. x_a (25.6MB) and aggr (25.6MB) both live in the 192MB L2,
// so this phase is bounded by L2 atomic throughput (E*H = 64M f32 atomics).
// ---------------------------------------------------------------------------
__global__ void metapath_scatter_kernel(const float* __restrict__ x,
                                        const int* __restrict__ e_dst,
                                        const int* __restrict__ e_src,
                                        float* __restrict__ aggr, int E) {
    int t = blockIdx.x * blockDim.x + threadIdx.x;
    int e = t >> 5;                 // wave32: one edge per wave
    if (e >= E) return;
    int lane = t & 31;              // lane covers 4 of the 128 features
    int dst = e_dst[e];
    int src = e_src[e];
    const float4 v = *(const float4*)(x + (size_t)src * H + lane * 4);
    float* d = aggr + (size_t)dst * H + lane * 4;
    atomicAdd(d + 0, v.x);
    atomicAdd(d + 1, v.y);
    atomicAdd(d + 2, v.z);
    atomicAdd(d + 3, v.w);
}

// ---------------------------------------------------------------------------
// Fused: h = relu(aggr @ Wl^T + x_a @ W01^T + bias_h); out = h @ Wout^T + bout
// Block = 128 threads (4 waves), 64 rows per block (4 row sub-tiles of 16).
// Each wave owns two 16-wide output column blocks for all 4 sub-tiles.
// K loop outermost so B fragments are loaded once and reused 4x:
//   per K-step: 4 B loads + 8 A loads, 16 v_wmma_f32_16x16x4_f32.
//
// f32 WMMA fragment layout (ISA 7.12.2):
//   A 16x4:  lane m (0-15): VGPR0=A[m][k0+0], VGPR1=A[m][k0+1]
//            lane m+16    : VGPR0=A[m][k0+2], VGPR1=A[m][k0+3]
//   B 4x16:  lane n (0-15): VGPR0=B[k0+0][n], VGPR1=B[k0+1][n]
//            lane n+16    : VGPR0=B[k0+2][n], VGPR1=B[k0+3][n]
//   C 16x16: VGPR r: lanes 0-15 -> (M=r, N=lane); lanes 16-31 -> (M=r+8)
// y = x @ W^T means B[k][n] = W[n*H + k].
// ---------------------------------------------------------------------------
__global__ __launch_bounds__(128) void metapath_gemm_kernel(
        const float* __restrict__ aggr,
        const float* __restrict__ x,
        const float* __restrict__ Wl,     // conv1_wl_w [H][H]
        const float* __restrict__ W01,    // folded conv1_w0_w + conv1_w1_w
        const float* __restrict__ biasH,  // folded bias
        const float* __restrict__ Wout,   // out_w [H][H]
        const float* __restrict__ bout,   // out_b
        float* __restrict__ out, int N) {
    __shared__ float lds_h[64 * H];       // 32KB intermediate h tile

    const int rowbase = blockIdx.x * 64;
    const int wave = threadIdx.x >> 5;
    const int lane = threadIdx.x & 31;
    const int l15  = lane & 15;
    const int hi   = lane >> 4;           // half-wave select (K pairs / M+8)

    const int col0 = (wave * 2 + 0) * 16;
    const int col1 = (wave * 2 + 1) * 16;

    // Per-sub-tile A row pointers (clamped so EXEC stays full for WMMA)
    const float* pA[4];
    const float* pX[4];
#pragma unroll
    for (int s = 0; s < 4; ++s) {
        int r = rowbase + s * 16 + l15;
        if (r > N - 1) r = N - 1;
        pA[s] = aggr + (size_t)r * H;
        pX[s] = x    + (size_t)r * H;
    }

    const float* __restrict__ pBl0 = Wl  + (size_t)(col0 + l15) * H;
    const float* __restrict__ pBl1 = Wl  + (size_t)(col1 + l15) * H;
    const float* __restrict__ pBx0 = W01 + (size_t)(col0 + l15) * H;
    const float* __restrict__ pBx1 = W01 + (size_t)(col1 + l15) * H;

    v8f c[4][2] = {};
    for (int kb = 0; kb < H; kb += 4) {
        const int k = kb + 2 * hi;
        v2f bl0 = { pBl0[k], pBl0[k + 1] };
        v2f bl1 = { pBl1[k], pBl1[k + 1] };
        v2f bx0 = { pBx0[k], pBx0[k + 1] };
        v2f bx1 = { pBx1[k], pBx1[k + 1] };
#pragma unroll
        for (int s = 0; s < 4; ++s) {
            v2f aA = { pA[s][k], pA[s][k + 1] };
            v2f aX = { pX[s][k], pX[s][k + 1] };
            c[s][0] = __builtin_amdgcn_wmma_f32_16x16x4_f32(false, aA, false, bl0, (short)0, c[s][0], false, false);
            c[s][1] = __builtin_amdgcn_wmma_f32_16x16x4_f32(false, aA, false, bl1, (short)0, c[s][1], false, false);
            c[s][0] = __builtin_amdgcn_wmma_f32_16x16x4_f32(false, aX, false, bx0, (short)0, c[s][0], false, false);
            c[s][1] = __builtin_amdgcn_wmma_f32_16x16x4_f32(false, aX, false, bx1, (short)0, c[s][1], false, false);
        }
    }

    // bias + ReLU -> LDS h tile (64 x 128)
    const float bh0 = biasH[col0 + l15];
    const float bh1 = biasH[col1 + l15];
#pragma unroll
    for (int s = 0; s < 4; ++s) {
#pragma unroll
        for (int r = 0; r < 8; ++r) {
            const int m = s * 16 + r + 8 * hi;
            float h0 = c[s][0][r] + bh0; h0 = h0 > 0.f ? h0 : 0.f;
            float h1 = c[s][1][r] + bh1; h1 = h1 > 0.f ? h1 : 0.f;
            lds_h[m * H + col0 + l15] = h0;
            lds_h[m * H + col1 + l15] = h1;
        }
    }
    __syncthreads();

    // out = h @ Wout^T + bout, same K-outer structure (A rows from LDS)
    const float* __restrict__ pBo0 = Wout + (size_t)(col0 + l15) * H;
    const float* __restrict__ pBo1 = Wout + (size_t)(col1 + l15) * H;
    v8f d[4][2] = {};
    for (int kb = 0; kb < H; kb += 4) {
        const int k = kb + 2 * hi;
        v2f b0 = { pBo0[k], pBo0[k + 1] };
        v2f b1 = { pBo1[k], pBo1[k + 1] };
#pragma unroll
        for (int s = 0; s < 4; ++s) {
            const float* hrow = &lds_h[(s * 16 + l15) * H];
            v2f aH = { hrow[k], hrow[k + 1] };
            d[s][0] = __builtin_amdgcn_wmma_f32_16x16x4_f32(false, aH, false, b0, (short)0, d[s][0], false, false);
            d[s][1] = __builtin_amdgcn_wmma_f32_16x16x4_f32(false, aH, false, b1, (short)0, d[s][1], false, false);
        }
    }

    const float bo0 = bout[col0 + l15];
    const float bo1 = bout[col1 + l15];
#pragma unroll
    for (int s = 0; s < 4; ++s) {
#pragma unroll
        for (int r = 0; r < 8; ++r) {
            const int m = rowbase + s * 16 + r + 8 * hi;
            if (m < N) {
                out[(size_t)m * H + col0 + l15] = d[s][0][r] + bo0;
                out[(size_t)m * H + col1 + l15] = d[s][1][r] + bo1;
            }
        }
    }
}

// ---------------------------------------------------------------------------
extern "C" void kernel_launch(void* const* d_in, const int* in_sizes, int n_in,
                              void* d_out, int out_size, void* d_ws, size_t ws_size,
                              hipStream_t stream) {
    // Inputs in setup_inputs() order; only the live h_a branch is needed
    // (h_b / conv0 / x_b / edge_ab are dead in the reference's return value).
    const float* x_a     = (const float*)d_in[0];
    const int*   edge_ba = (const int*)  d_in[3];   // [2, E] flat
    const float* c1_w0_w = (const float*)d_in[10];
    const float* c1_w0_b = (const float*)d_in[11];
    const float* c1_wl_w = (const float*)d_in[12];
    const float* c1_wl_b = (const float*)d_in[13];
    const float* c1_w1_w = (const float*)d_in[14];
    const float* c1_w1_b = (const float*)d_in[15];
    const float* out_w   = (const float*)d_in[16];
    const float* out_b   = (const float*)d_in[17];

    const int N = in_sizes[0] / H;
    const int E = in_sizes[3] / 2;

    // Workspace layout: aggr[N*H] | W01[H*H] | biasH[H]
    float* aggr  = (float*)d_ws;
    float* W01   = aggr + (size_t)N * H;
    float* biasH = W01 + H * H;

    // (a) fold weights/biases
    metapath_prep_kernel<<<(H * H + 255) / 256, 256, 0, stream>>>(
        c1_w0_w, c1_w1_w, c1_w0_b, c1_wl_b, c1_w1_b, W01, biasH);

    // (b) zero aggregation buffer
    const int n4 = (N * H) / 4;
    metapath_zero_kernel<<<(n4 + 255) / 256, 256, 0, stream>>>((float4*)aggr, n4);

    // (c) scatter-add: one wave32 per edge
    const long long sthreads = (long long)E * 32;
    metapath_scatter_kernel<<<(int)((sthreads + 255) / 256), 256, 0, stream>>>(
        x_a, edge_ba /*dst = edge[0]*/, edge_ba + E /*src = edge[1]*/, aggr, E);

    // (d) fused double-GEMM + bias + relu + output GEMM (64 rows / block)
    metapath_gemm_kernel<<<(N + 63) / 64, 128, 0, stream>>>(
        aggr, x_a, c1_wl_w, W01, biasH, out_w, out_b, (float*)d_out, N);
}